// Transform_42872363549171
// MI455X (gfx1250) — compile-verified
//
#include <hip/hip_runtime.h>

typedef __attribute__((ext_vector_type(2))) float v2f;
typedef __attribute__((ext_vector_type(8))) float v8f;

#define B_      16
#define N_      2048
#define D_      3
#define T_      64
#define R_      64
#define NSPLIT  16
#define NPW     (N_ / NSPLIT)   // 128 n-values per wave

// sigmoid(z) = 0.5 + 0.5*tanh(z/2); we accumulate raw tanh and fold the 0.5s.
__device__ __forceinline__ float fast_tanh(float x) {
#if __has_builtin(__builtin_amdgcn_tanhf)
    return __builtin_amdgcn_tanhf(x);          // v_tanh_f32 (gfx1250 TRANS op)
#else
    float e = __builtin_amdgcn_exp2f(x * -2.8853900817779268f);
    float r = __builtin_amdgcn_rcpf(1.0f + e);
    return __builtin_fmaf(2.0f, r, -1.0f);
#endif
}

// out[b,r,t] starts at 0.5*N (the constant half of every sigmoid).
__global__ __launch_bounds__(256) void ect_init_kernel(float* __restrict__ out) {
    int i = blockIdx.x * blockDim.x + threadIdx.x;
    out[i] = 0.5f * (float)N_;   // 1024.0f
}

__global__ __launch_bounds__(256) void ect_kernel(
    const float* __restrict__ x,      // [B, N, 3]
    const float* __restrict__ v,      // [3, T]
    const float* __restrict__ lin,    // [R]
    const int*   __restrict__ scale_p,
    float* __restrict__ out)          // [B, R, T]
{
    const float scale = (float)scale_p[0];
    const float kh    = 0.5f * scale;          // arg = kh*lin[r] - kh*nh

    const int lane  = threadIdx.x & 31;
    const int wave  = threadIdx.x >> 5;        // 0..7
    const int t_blk = wave & 3;                // 4 t-blocks of 16
    const int s_bit = wave >> 2;               // 0..1

    const int bid   = blockIdx.x;              // 0..511
    const int b     = bid >> 5;                // 0..15
    const int rem   = bid & 31;
    const int r_blk = rem >> 3;                // 4 r-blocks of 16
    const int s_hi  = rem & 7;                 // 0..7
    const int s     = s_hi * 2 + s_bit;        // n-split 0..15
    const int n0w   = s * NPW;

    const int tl = lane & 15;                  // col within tile
    const int hi = lane >> 4;                  // 0: K0/K1 half, 1: K2/K3 half
    const int t  = t_blk * 16 + tl;

    // ---- B operand (v tile, 4x16 f32, K=3 padded to 4), loaded once ----
    // ISA layout: VGPR0 = {lanes0-15: K0 | lanes16-31: K2}, VGPR1 = {K1 | K3(=0)}
    float b0 = v[(hi * 2) * T_ + t];           // d = 0 or d = 2
    float b1 = v[1 * T_ + t];                  // d = 1 (always a valid address)
    v2f bmat;
    bmat.x = b0;
    bmat.y = hi ? 0.0f : b1;                   // pad K=3 with zero

    // ---- per-r thresholds (wave-uniform scalar loads -> SGPRs) ----
    float klr[16];
#pragma unroll
    for (int rr = 0; rr < 16; ++rr)
        klr[rr] = kh * lin[r_blk * 16 + rr];

    float acc[16];
#pragma unroll
    for (int rr = 0; rr < 16; ++rr) acc[rr] = 0.0f;

    const float* xb = x + (size_t)b * N_ * D_;
    const int d_sel = hi ? 2 : 0;              // which x component this half loads

    // ---- software-pipelined x loads: fetch chunk 0 ----
    size_t base = (size_t)(n0w + tl) * D_;
    float a0 = xb[base + d_sel];
    float a1 = xb[base + 1];

    // ---- main loop: 8 chunks of 16 points ----
    for (int c = 0; c < NPW; c += 16) {
        v2f amat;
        amat.x = a0;
        amat.y = hi ? 0.0f : a1;               // pad K=3 with zero

        // Prefetch next chunk's A operand (uniform clamp keeps address valid,
        // so no EXEC divergence; results consumed next iteration, hiding L2
        // latency behind this iteration's tanh block).
        const int cn = (c + 16 < NPW) ? (c + 16) : c;
        const size_t basen = (size_t)(n0w + cn + tl) * D_;
        a0 = xb[basen + d_sel];
        a1 = xb[basen + 1];

        v8f czero = {};
        // nh tile: D[m, t] = sum_d x[n0+m, d] * v[d, t]   (16n x 16t)
        v8f nh = __builtin_amdgcn_wmma_f32_16x16x4_f32(
            false, amat, false, bmat, (short)0, czero, false, false);

        float h[8];
#pragma unroll
        for (int j = 0; j < 8; ++j) h[j] = nh[j];   // kh folds into the FMA below

        // 16 r-levels x 8 rows: exactly 1 v_fma + 1 v_tanh + 1 v_add each
#pragma unroll
        for (int rr = 0; rr < 16; ++rr) {
            float a = fast_tanh(__builtin_fmaf(-kh, h[0], klr[rr]));
#pragma unroll
            for (int j = 1; j < 8; ++j)
                a += fast_tanh(__builtin_fmaf(-kh, h[j], klr[rr]));
            acc[rr] += a;
        }
    }

    // ---- combine partials: out += 0.5 * sum(tanh) ----
    float* outb = out + ((size_t)b * R_ + r_blk * 16) * T_ + t;
#pragma unroll
    for (int rr = 0; rr < 16; ++rr) {
#if defined(__HIP_DEVICE_COMPILE__)
        unsafeAtomicAdd(outb + rr * T_, 0.5f * acc[rr]);   // global_atomic_add_f32
#else
        atomicAdd(outb + rr * T_, 0.5f * acc[rr]);
#endif
    }
}

extern "C" void kernel_launch(void* const* d_in, const int* in_sizes, int n_in,
                              void* d_out, int out_size, void* d_ws, size_t ws_size,
                              hipStream_t stream) {
    const float* x     = (const float*)d_in[0];   // [16, 2048, 3]
    const float* v     = (const float*)d_in[1];   // [3, 64]
    const float* lin   = (const float*)d_in[2];   // [64]
    const int*   scale = (const int*)d_in[3];     // scalar 500
    float* out = (float*)d_out;                   // [16, 64, 64] = 65536 floats

    // Initialize output with the folded-out constant 0.5*N per element.
    ect_init_kernel<<<dim3(out_size / 256), dim3(256), 0, stream>>>(out);

    // 512 blocks x 8 waves = 4096 wave-tasks over (b, r_blk, t_blk, n_split).
    ect_kernel<<<dim3(512), dim3(256), 0, stream>>>(x, v, lin, scale, out);
}